// VNMSparseLinear_62302795596652
// MI455X (gfx1250) — compile-verified
//
#include <hip/hip_runtime.h>

typedef __bf16 bf16;
typedef __attribute__((ext_vector_type(16))) __bf16       v16bf;
typedef __attribute__((ext_vector_type(32))) __bf16       v32bf;
typedef __attribute__((ext_vector_type(8)))  float        v8f;
typedef __attribute__((ext_vector_type(4)))  unsigned int u32x4;
typedef __attribute__((ext_vector_type(4)))  float        f32x4;

#define D_IN  4096   // in_features
#define OUTF  4096   // out_features
#define NROWS 8192   // B*S = 4*2048
#define NBLK  512    // D_IN / 8
#define KGATH 2048   // NBLK * 4 live columns (top-4 per 8-block, shared across rows)
#define KSTOR 1024   // KGATH / 2 (2:4 compacted)
#define NGRP  512    // KGATH / 4 groups per row
#define IWORDS 64    // NGRP / 8 packed index u32 per row

#define SBK  64                 // expanded K per swmmac step
#define SNKB (KGATH / SBK)      // 32 K-steps

union FragA { u32x4 q[2]; v16bf v; };
union FragB { u32x4 q[4]; v32bf v; };

// ---------------------------------------------------------------------------
// Kernel 1: per-8-column-block importance = sum_o |W[o,c]|, pick top-4 cols.
// ---------------------------------------------------------------------------
__global__ __launch_bounds__(256) void k_topk(const float* __restrict__ W,
                                              int* __restrict__ g)
{
    __shared__ float s_red[256];
    __shared__ float s_imp[8];
    const int blk = blockIdx.x;
    const int tid = threadIdx.x;
    const int c0  = blk * 8;

    float acc[8];
#pragma unroll
    for (int j = 0; j < 8; ++j) acc[j] = 0.0f;

    for (int o = tid; o < OUTF; o += 256) {
        const f32x4* p = (const f32x4*)(W + (size_t)o * D_IN + c0);
        f32x4 w0 = p[0], w1 = p[1];
        acc[0] += fabsf(w0[0]); acc[1] += fabsf(w0[1]);
        acc[2] += fabsf(w0[2]); acc[3] += fabsf(w0[3]);
        acc[4] += fabsf(w1[0]); acc[5] += fabsf(w1[1]);
        acc[6] += fabsf(w1[2]); acc[7] += fabsf(w1[3]);
    }

    for (int j = 0; j < 8; ++j) {
        s_red[tid] = acc[j];
        __syncthreads();
        for (int s = 128; s > 0; s >>= 1) {
            if (tid < s) s_red[tid] += s_red[tid + s];
            __syncthreads();
        }
        if (tid == 0) s_imp[j] = s_red[0];
        __syncthreads();
    }

    if (tid == 0) {
        float v[8]; bool used[8];
#pragma unroll
        for (int j = 0; j < 8; ++j) { v[j] = s_imp[j]; used[j] = false; }
        for (int i = 0; i < 4; ++i) {          // descending, first index wins ties
            int best = -1; float bv = -1.0f;
            for (int j = 0; j < 8; ++j)
                if (!used[j] && v[j] > bv) { bv = v[j]; best = j; }
            used[best] = true;
            g[blk * 4 + i] = c0 + best;
        }
    }
}

// ---------------------------------------------------------------------------
// Kernel 2: soft-threshold + 2:4 compaction.
//   Wc2[o][k']  : 2 kept values per group of 4 (ascending position order)
//   Widx[o][w]  : 8 groups per u32, nibble = idx0 | (idx1<<2), idx0 < idx1
//   pnum/pden   : per-row partials for beta
// ---------------------------------------------------------------------------
__global__ __launch_bounds__(256) void k_build2(const float* __restrict__ W,
                                                const int*   __restrict__ g,
                                                bf16*        __restrict__ Wc2,
                                                unsigned int* __restrict__ Widx,
                                                float*       __restrict__ pnum,
                                                float*       __restrict__ pden)
{
    __shared__ unsigned char s_code[NGRP];
    __shared__ float s_n[256];
    __shared__ float s_d[256];
    const int o   = blockIdx.x;
    const int tid = threadIdx.x;

    float num = 0.0f, den = 0.0f;
    for (int gq = tid; gq < NGRP; gq += 256) {
        float w[4];
#pragma unroll
        for (int i = 0; i < 4; ++i)
            w[i] = W[(size_t)o * D_IN + g[gq * 4 + i]];

        const float a0 = fabsf(w[0]), a1 = fabsf(w[1]);
        const float a2 = fabsf(w[2]), a3 = fabsf(w[3]);
        const float lo1 = fminf(a0, a1), hi1 = fmaxf(a0, a1);
        const float lo2 = fminf(a2, a3), hi2 = fmaxf(a2, a3);
        const float thr = fminf(fmaxf(lo1, lo2), fminf(hi1, hi2)); // 2nd smallest

        float sv[4];
#pragma unroll
        for (int i = 0; i < 4; ++i) {
            sv[i] = copysignf(fmaxf(fabsf(w[i]) - thr, 0.0f), w[i]);
            num += w[i] * sv[i];
            den += sv[i] * sv[i];
        }

        // positions of the two largest |w| (first index wins ties)
        int i0 = 0;
        for (int j = 1; j < 4; ++j) if (fabsf(w[j]) > fabsf(w[i0])) i0 = j;
        int i1 = -1;
        for (int j = 0; j < 4; ++j)
            if (j != i0 && (i1 < 0 || fabsf(w[j]) > fabsf(w[i1]))) i1 = j;
        const int lo = (i0 < i1) ? i0 : i1;
        const int hi = (i0 < i1) ? i1 : i0;

        Wc2[(size_t)o * KSTOR + gq * 2 + 0] = (bf16)sv[lo];
        Wc2[(size_t)o * KSTOR + gq * 2 + 1] = (bf16)sv[hi];
        s_code[gq] = (unsigned char)(lo | (hi << 2));
    }

    __syncthreads();
    if (tid < IWORDS) {
        unsigned int word = 0;
#pragma unroll
        for (int b = 0; b < 8; ++b)
            word |= ((unsigned int)s_code[tid * 8 + b]) << (b * 4);
        Widx[(size_t)o * IWORDS + tid] = word;
    }

    s_n[tid] = num; s_d[tid] = den;
    __syncthreads();
    for (int s = 128; s > 0; s >>= 1) {
        if (tid < s) { s_n[tid] += s_n[tid + s]; s_d[tid] += s_d[tid + s]; }
        __syncthreads();
    }
    if (tid == 0) { pnum[o] = s_n[0]; pden[o] = s_d[0]; }
}

// ---------------------------------------------------------------------------
// Kernel 3: deterministic final reduction -> beta scalar.
// ---------------------------------------------------------------------------
__global__ __launch_bounds__(256) void k_beta(const float* __restrict__ pnum,
                                              const float* __restrict__ pden,
                                              float* __restrict__ beta)
{
    __shared__ float s_n[256];
    __shared__ float s_d[256];
    const int tid = threadIdx.x;
    float n = 0.0f, d = 0.0f;
    for (int i = tid; i < OUTF; i += 256) { n += pnum[i]; d += pden[i]; }
    s_n[tid] = n; s_d[tid] = d;
    __syncthreads();
    for (int s = 128; s > 0; s >>= 1) {
        if (tid < s) { s_n[tid] += s_n[tid + s]; s_d[tid] += s_d[tid + s]; }
        __syncthreads();
    }
    if (tid == 0) beta[0] = (s_d[0] > 0.0f) ? (s_n[0] / s_d[0]) : 1.0f;
}

// ---------------------------------------------------------------------------
// Kernel 4: gather live columns of x -> xg (bf16, 8192 x 2048).
// ---------------------------------------------------------------------------
__global__ __launch_bounds__(256) void k_gather(const float* __restrict__ x,
                                                const int*   __restrict__ g,
                                                bf16*        __restrict__ xg)
{
    const int idx = blockIdx.x * 256 + threadIdx.x;
    const int s   = idx >> 9;
    const int k0  = (idx & 511) * 4;
    const float* xr = x + (size_t)s * D_IN;
    union { bf16 h[4]; unsigned long long u; } pk;
#pragma unroll
    for (int i = 0; i < 4; ++i) pk.h[i] = (bf16)xr[g[k0 + i]];
    *(unsigned long long*)(xg + (size_t)s * KGATH + k0) = pk.u;
}

// ---------------------------------------------------------------------------
// Kernel 5: sparse GEMM via V_SWMMAC_F32_16X16X64_BF16 with ASYNC global->LDS
// tile feed (GLOBAL_LOAD_ASYNC_TO_LDS_B128, ASYNCcnt).
//   A (sparse) = Wc2 rows (M = o), B (dense) = xg columns (N = s).
//   128(o) x 128(s) tile, expanded K-step 64 (stored 32), double-buffered LDS.
//   Each wave: 32(o) x 64(s) region = 2x4 swmmac subtiles.
// ---------------------------------------------------------------------------
__global__ __launch_bounds__(256) void k_gemm_sp(const bf16*  __restrict__ Wc2,
                                                 const unsigned int* __restrict__ Widx,
                                                 const bf16*  __restrict__ xg,
                                                 const float* __restrict__ betap,
                                                 float*       __restrict__ out)
{
    __shared__ __attribute__((aligned(16))) bf16 As[2][128 * 32]; // stored-K, 8KB each
    __shared__ __attribute__((aligned(16))) bf16 Bs[2][128 * 64]; // expanded-K, 16KB each

    const int tid  = threadIdx.x;
    const int lane = tid & 31;        // wave32
    const int wave = tid >> 5;
    const int o0   = blockIdx.y * 128;
    const int s0   = blockIdx.x * 128;

    // tile loaders: 2 threads per row; per-thread region is contiguous in both
    // global and LDS, so one LDS-addr VGPR + inst offsets feed the async DMA.
    const int lrow = tid >> 1;        // 0..127
    const int lhal = tid & 1;
    const bf16* gA = Wc2 + (size_t)(o0 + lrow) * KSTOR + lhal * 16; // 32B per step
    const bf16* gB = xg  + (size_t)(s0 + lrow) * KGATH + lhal * 32; // 64B per step

    const unsigned int ldsA0 = (unsigned int)(uintptr_t)&As[0][lrow * 32 + lhal * 16];
    const unsigned int ldsB0 = (unsigned int)(uintptr_t)&Bs[0][lrow * 64 + lhal * 32];

    const int rbase = (wave & 3) * 32;   // o within tile
    const int cbase = (wave >> 2) * 64;  // s within tile
    const int frow  = lane & 15;
    const int fsec  = lane >> 4;

    // sparse-index pointers: word (kb*2 + fsec) of row (o0 + rbase + i*16 + frow)
    const unsigned int* gI0 = Widx + (size_t)(o0 + rbase + frow) * IWORDS + fsec;
    const unsigned int* gI1 = Widx + (size_t)(o0 + rbase + 16 + frow) * IWORDS + fsec;

    v8f acc[2][4] = {};

    // prologue: async DMA tile 0 -> buf 0 (A: 2x b128, B: 4x b128 per thread)
    {
        unsigned long long ga = (unsigned long long)(uintptr_t)gA;
        unsigned long long gb = (unsigned long long)(uintptr_t)gB;
        asm volatile(
            "global_load_async_to_lds_b128 %0, %1, off\n\t"
            "global_load_async_to_lds_b128 %0, %1, off offset:16\n\t"
            "global_load_async_to_lds_b128 %2, %3, off\n\t"
            "global_load_async_to_lds_b128 %2, %3, off offset:16\n\t"
            "global_load_async_to_lds_b128 %2, %3, off offset:32\n\t"
            "global_load_async_to_lds_b128 %2, %3, off offset:48"
            :: "v"(ldsA0), "v"(ga), "v"(ldsB0), "v"(gb) : "memory");
    }

    int buf = 0;
    for (int kb = 0; kb < SNKB; ++kb) {
        // my async writes into `buf` must land before anyone reads it
        asm volatile("s_wait_asynccnt 0" ::: "memory");
        __syncthreads();   // also waits DScnt -> all reads of `nb` (kb-1) done

        const bool more = (kb + 1) < SNKB;
        if (more) {
            const int nb = buf ^ 1;
            const unsigned int ldsA = ldsA0 + nb * (128 * 32 * 2);
            const unsigned int ldsB = ldsB0 + nb * (128 * 64 * 2);
            unsigned long long ga = (unsigned long long)(uintptr_t)(gA + (kb + 1) * 32);
            unsigned long long gb = (unsigned long long)(uintptr_t)(gB + (kb + 1) * 64);
            asm volatile(
                "global_load_async_to_lds_b128 %0, %1, off\n\t"
                "global_load_async_to_lds_b128 %0, %1, off offset:16\n\t"
                "global_load_async_to_lds_b128 %2, %3, off\n\t"
                "global_load_async_to_lds_b128 %2, %3, off offset:16\n\t"
                "global_load_async_to_lds_b128 %2, %3, off offset:32\n\t"
                "global_load_async_to_lds_b128 %2, %3, off offset:48"
                :: "v"(ldsA), "v"(ga), "v"(ldsB), "v"(gb) : "memory");
            if (kb + 2 < SNKB) {
                __builtin_prefetch(gA + (kb + 2) * 32, 0, 3);
                __builtin_prefetch(gB + (kb + 2) * 64, 0, 3);
            }
        }

        // sparse index words (1 VGPR each): lane r -> groups K[0..31] of row r,
        // lane r+16 -> groups K[32..63]
        int ai[2];
        ai[0] = (int)gI0[kb * 2];
        ai[1] = (int)gI1[kb * 2];

        // A fragments (stored 16x32 bf16, standard 16-bit A layout)
        FragA a[2];
#pragma unroll
        for (int i = 0; i < 2; ++i) {
            const int r = rbase + i * 16 + frow;
            const u32x4* p = (const u32x4*)&As[buf][r * 32];
            a[i].q[0] = p[fsec];
            a[i].q[1] = p[2 + fsec];
        }
        // B fragments (64x16 bf16): lane=col, VGPRs 0..7 = K[fsec*16..+15],
        // VGPRs 8..15 = K[32+fsec*16..+15]
        FragB b[4];
#pragma unroll
        for (int j = 0; j < 4; ++j) {
            const int c = cbase + j * 16 + frow;
            const u32x4* p = (const u32x4*)&Bs[buf][c * 64];
            b[j].q[0] = p[fsec * 2];
            b[j].q[1] = p[fsec * 2 + 1];
            b[j].q[2] = p[4 + fsec * 2];
            b[j].q[3] = p[5 + fsec * 2];
        }

#pragma unroll
        for (int i = 0; i < 2; ++i)
#pragma unroll
            for (int j = 0; j < 4; ++j)
                asm volatile("v_swmmac_f32_16x16x64_bf16 %0, %1, %2, %3"
                             : "+v"(acc[i][j])
                             : "v"(a[i].v), "v"(b[j].v), "v"(ai[i]));
        // SWMMAC->VALU/LDS hazard cover (1 NOP + 2 coexec per ISA 7.12.1);
        // inline asm bypasses the compiler's automatic insertion.
        asm volatile("s_nop 3" ::: "memory");

        buf ^= 1;
    }

    // epilogue: D element (m,n): o = o0+rbase+i*16 + vg + 8*fsec (contiguous in vg),
    // s = s0+cbase+j*16+frow  ->  two b128 stores per subtile per lane.
    const float beta = betap[0];
#pragma unroll
    for (int i = 0; i < 2; ++i) {
        const int obase = o0 + rbase + i * 16 + fsec * 8;
#pragma unroll
        for (int j = 0; j < 4; ++j) {
            const int srow = s0 + cbase + j * 16 + frow;
            union { v8f v; f32x4 f[2]; } r;
            r.v = acc[i][j] * beta;
            f32x4* po = (f32x4*)(out + (size_t)srow * OUTF + obase);
            po[0] = r.f[0];
            po[1] = r.f[1];
        }
    }
}

// ---------------------------------------------------------------------------
// Workspace layout (bytes):
//   0       : int   g[2048]
//   8192    : float pnum[4096]
//   24576   : float pden[4096]
//   40960   : float beta[1]
//   65536   : bf16  Wc2[4096*1024]     (8 MB, 2:4 compacted)
//   8454144 : u32   Widx[4096*64]      (1 MB, packed 2-bit index pairs)
//   9502720 : bf16  xg[8192*2048]      (32 MB)
// ---------------------------------------------------------------------------
extern "C" void kernel_launch(void* const* d_in, const int* in_sizes, int n_in,
                              void* d_out, int out_size, void* d_ws, size_t ws_size,
                              hipStream_t stream)
{
    const float* x = (const float*)d_in[0];
    const float* W = (const float*)d_in[1];
    float* out = (float*)d_out;

    char*  ws   = (char*)d_ws;
    int*   g    = (int*)(ws + 0);
    float* pnum = (float*)(ws + 8192);
    float* pden = (float*)(ws + 24576);
    float* beta = (float*)(ws + 40960);
    bf16*  Wc2  = (bf16*)(ws + 65536);
    unsigned int* Widx = (unsigned int*)(ws + 65536 + (size_t)OUTF * KSTOR * 2);
    bf16*  xg   = (bf16*)(ws + 65536 + (size_t)OUTF * KSTOR * 2
                             + (size_t)OUTF * IWORDS * 4);

    k_topk   <<<NBLK, 256, 0, stream>>>(W, g);
    k_build2 <<<OUTF, 256, 0, stream>>>(W, g, Wc2, Widx, pnum, pden);
    k_beta   <<<1, 256, 0, stream>>>(pnum, pden, beta);
    k_gather <<<(NROWS * (KGATH / 4)) / 256, 256, 0, stream>>>(x, g, xg);
    k_gemm_sp<<<dim3(NROWS / 128, OUTF / 128), 256, 0, stream>>>(Wc2, Widx, xg, beta, out);
}